// HmmCell_13451837571298
// MI455X (gfx1250) — compile-verified
//
#include <hip/hip_runtime.h>
#include <cstdint>
#include <cstddef>

// ---------------------------------------------------------------------------
// HMM scaled forward algorithm on MI455X (gfx1250).
//
// M=2 models, B=64 batch, Q=512 states, L=512 steps.
// emission:(L,M,B,Q) f32, A:(M,Q,Q) f32, init:(M,Q) f32, out ll:(M,B) f32.
//
// 8 persistent workgroups, one per (model, 16-batch tile) chain; the scan is
// sequential so the only sync-free parallelism is across (model, batch).
// Per step: acc = u_{t-1} @ A via V_WMMA_F32_16X16X4_F32; R = acc * (1/S),
// u_t = max(E,eps)*max(R,eps) stored raw into double-buffered LDS sf;
// after a barrier each wave row-sums its own batch row (b128 LDS loads +
// 5-level butterfly) -> S[w], ll_w += log(S_w). Emission tiles stream in
// via Tensor Data Mover, double buffered, overlapped with the GEMM.
// ---------------------------------------------------------------------------

typedef float    v2f __attribute__((ext_vector_type(2)));
typedef float    v4f __attribute__((ext_vector_type(4)));
typedef float    v8f __attribute__((ext_vector_type(8)));
typedef unsigned v4u __attribute__((ext_vector_type(4)));
typedef int      v8i __attribute__((ext_vector_type(8)));
typedef int      v4i __attribute__((ext_vector_type(4)));

#define EPS_F     1e-16f
#define M_MODELS  2
#define B_BATCH   64
#define Q_STATES  512
#define L_STEPS   512
// sf row stride padded: stride 516 dwords -> the 16x4 WMMA A-fragment
// ds_load_b64 pattern hits 64 distinct banks (bank = (4*row + k) % 64).
#define SFSTR     516

#if __has_builtin(__builtin_amdgcn_tensor_load_to_lds)
#define USE_TDM 1
#else
#define USE_TDM 0
#endif

__device__ __forceinline__ unsigned lds_byte_off(const void* p) {
  // LDS aperture: flat addr low 32 bits are the LDS byte address.
  return (unsigned)(uintptr_t)p;
}

#if USE_TDM
// 2D TDM load: tile_dim0 = 512 f32 (contiguous Q), tile_dim1 = 16 rows
// (batch), tensor_dim0_stride = 512 f32. D# per CDNA5 ISA.
__device__ __forceinline__ void tdm_load_tile(const float* g, unsigned lds) {
  unsigned long long ga = (unsigned long long)(uintptr_t)g;
  v4u g0 = {
    1u,                                                   // count=1, user mode
    lds,                                                  // lds_addr (bytes)
    (unsigned)ga,                                         // global_addr[31:0]
    (unsigned)((ga >> 32) & 0x1FFFFFFull) | 0x80000000u   // addr[56:32] | type=2
  };
  v8i g1 = {
    0x00020000,            // workgroup_mask=0, data_size=2 (4B)
    (int)(512u << 16),     // tensor_dim0[15:0]=512 at bits[63:48]
    (int)(16u  << 16),     // tensor_dim0 hi=0, tensor_dim1[15:0]=16
    (int)(512u << 16),     // tensor_dim1 hi=0, tile_dim0=512
    16,                    // tile_dim1=16, tile_dim2=0
    512,                   // tensor_dim0_stride[31:0]=512
    0, 0                   // stride hi / dim1_stride = 0
  };
  v4i z4 = {0, 0, 0, 0};
  v8i z8 = {0, 0, 0, 0, 0, 0, 0, 0};
  __builtin_amdgcn_tensor_load_to_lds(g0, g1, z4, z4, z8, 0);
}
#endif

// Sum one 512-float LDS row: 16 floats/lane via 4x ds_load_b128, local VALU
// sum, then a 5-level butterfly. Every lane returns the full row sum.
__device__ __forceinline__ float row_sum_512(const float* rowp, int l) {
  const float* p = rowp + l * 16;
  v4f a = *(const v4f*)(p + 0);
  v4f b = *(const v4f*)(p + 4);
  v4f c = *(const v4f*)(p + 8);
  v4f d = *(const v4f*)(p + 12);
  v4f s = (a + b) + (c + d);
  float red = (s.x + s.y) + (s.z + s.w);
  red += __shfl_xor(red, 1);
  red += __shfl_xor(red, 2);
  red += __shfl_xor(red, 4);
  red += __shfl_xor(red, 8);
  red += __shfl_xor(red, 16);
  return red;
}

__global__ __launch_bounds__(512, 1) void hmm_forward_kernel(
    const float* __restrict__ Eall,   // (L, M, B, Q)
    const float* __restrict__ Aall,   // (M, Q, Q)
    const float* __restrict__ initd,  // (M, Q)
    float* __restrict__ out) {        // (M, B)
  // Double-buffered unnormalized forward variable u (16 rows x 512 states).
  __shared__ __align__(16) float sfb[2][16 * SFSTR];
#if USE_TDM
  __shared__ __align__(16) float Eb[2][16 * Q_STATES];    // emission dbl-buf
#endif
  __shared__ __align__(16) float Sbuf[16];                // per-row sums

  const int tid = threadIdx.x;
  const int l   = tid & 31;      // lane in wave32
  const int lr  = l & 15;        // lane % 16
  const int lh  = l >> 4;        // lane half (0/1)
  const int wv  = tid >> 5;      // wave id, 0..15 (also owns batch row wv)

  const int m  = blockIdx.x >> 2;          // model
  const int b0 = (blockIdx.x & 3) << 4;    // batch tile base

  const float* Am    = Aall + (size_t)m * Q_STATES * Q_STATES;
  const float* Ebase = Eall + ((size_t)m * B_BATCH + b0) * Q_STATES;
  const size_t Estep = (size_t)M_MODELS * B_BATCH * Q_STATES;  // per-t stride

#if USE_TDM
  if (wv == 0) {
    tdm_load_tile(Ebase, lds_byte_off(&Eb[0][0]));
    tdm_load_tile(Ebase + Estep, lds_byte_off(&Eb[1][0]));
    __builtin_amdgcn_s_wait_tensorcnt(1);   // E[0] landed
  }
#endif
  __syncthreads();

  // --------------------------- init step (t = 0) ---------------------------
  // u_0 = max(E0,eps) * max(init,eps), stored RAW into sfb[1]
  // (step t reads sfb[t&1], writes sfb[(t+1)&1]).
  {
#if USE_TDM
    const float* E0 = &Eb[0][0];
#else
    const float* E0 = Ebase;
#endif
    const float iv = fmaxf(initd[m * Q_STATES + tid], EPS_F);
#pragma unroll
    for (int i = 0; i < 16; ++i)
      sfb[1][i * SFSTR + tid] = fmaxf(E0[i * Q_STATES + tid], EPS_F) * iv;
  }
  __syncthreads();

  float llv;
  {
    float red = row_sum_512(&sfb[1][wv * SFSTR], l);
    llv = logf(red);
    if (l == 0) Sbuf[wv] = red;
  }
#if USE_TDM
  if (wv == 0) {
    tdm_load_tile(Ebase + 2 * Estep, lds_byte_off(&Eb[0][0]));  // prefetch E[2]
    __builtin_amdgcn_s_wait_tensorcnt(1);                       // E[1] landed
  }
#endif
  __syncthreads();

  // ------------------------------ main scan --------------------------------
  const int n0a = wv << 4;         // first N tile for this wave
  const int n0b = n0a + 256;       // second N tile

  for (int t = 1; t < L_STEPS; ++t) {
    const int p = t & 1;
    const float* cur = &sfb[p][0];        // u_{t-1} (raw)
    float*       nxt = &sfb[1 - p][0];    // u_t destination

    // 1/S_{t-1}: two b128 LDS loads + 8 v_rcp, overlaps the GEMM below.
    v4f sLo = *(const v4f*)(Sbuf + (lh << 3));
    v4f sHi = *(const v4f*)(Sbuf + (lh << 3) + 4);
    float inv[8];
#pragma unroll
    for (int j = 0; j < 4; ++j) {
      inv[j]     = __builtin_amdgcn_rcpf(sLo[j]);
      inv[4 + j] = __builtin_amdgcn_rcpf(sHi[j]);
    }

    // [A] acc = u_{t-1} @ A : K loop of V_WMMA_F32_16X16X4_F32, 2 N tiles.
    v8f acc0 = {0.f, 0.f, 0.f, 0.f, 0.f, 0.f, 0.f, 0.f};
    v8f acc1 = {0.f, 0.f, 0.f, 0.f, 0.f, 0.f, 0.f, 0.f};
    {
      const float* sfrow = cur + lr * SFSTR + 2 * lh;  // WMMA-A fragment base
      const float* bp = Am + (size_t)(2 * lh) * Q_STATES + lr;
#pragma unroll 8
      for (int k = 0; k < Q_STATES; k += 4) {
        v2f av = *(const v2f*)(sfrow + k);        // A frag: rows=batch, K=q
        const float* b = bp + (size_t)k * Q_STATES;
        v2f bv0 = { b[n0a], b[Q_STATES + n0a] };  // B frag: K rows, N cols
        v2f bv1 = { b[n0b], b[Q_STATES + n0b] };
        acc0 = __builtin_amdgcn_wmma_f32_16x16x4_f32(
            false, av, false, bv0, (short)0, acc0, false, false);
        acc1 = __builtin_amdgcn_wmma_f32_16x16x4_f32(
            false, av, false, bv1, (short)0, acc1, false, false);
      }
    }

    // [B] R = acc/S_{t-1}; u_t = max(E,eps)*max(R,eps) stored raw into nxt.
#if USE_TDM
    const float* Ec = &Eb[p][0];
#else
    const float* Ec = Ebase + (size_t)t * Estep;
#endif
#pragma unroll
    for (int i = 0; i < 8; ++i) {
      const int row = i + (lh << 3);   // C/D layout: vgpr i, halves +8
      float r0 = fmaxf(acc0[i] * inv[i], EPS_F);
      float r1 = fmaxf(acc1[i] * inv[i], EPS_F);
      float e0 = fmaxf(Ec[row * Q_STATES + n0a + lr], EPS_F);
      float e1 = fmaxf(Ec[row * Q_STATES + n0b + lr], EPS_F);
      nxt[row * SFSTR + n0a + lr] = e0 * r0;
      nxt[row * SFSTR + n0b + lr] = e1 * r1;
    }
    __syncthreads();   // [C] u_t complete; all cur/E/S reads done

    // [D] wave wv reduces its own batch row: S[wv], ll += log(S).
    {
      float red = row_sum_512(nxt + wv * SFSTR, l);
      llv += logf(red);
      if (l == 0) Sbuf[wv] = red;
    }
#if USE_TDM
    if (wv == 0) {
      if (t + 2 < L_STEPS) {
        tdm_load_tile(Ebase + (size_t)(t + 2) * Estep,
                      lds_byte_off(&Eb[p][0]));        // buf p free after [C]
        __builtin_amdgcn_s_wait_tensorcnt(1);          // E[t+1] landed
      } else {
        __builtin_amdgcn_s_wait_tensorcnt(0);
      }
    }
#endif
    __syncthreads();   // [E] S + u_t ready for next step
  }

  if (l == 0) out[m * B_BATCH + b0 + wv] = llv;
}

extern "C" void kernel_launch(void* const* d_in, const int* in_sizes, int n_in,
                              void* d_out, int out_size, void* d_ws, size_t ws_size,
                              hipStream_t stream) {
  (void)in_sizes; (void)n_in; (void)out_size; (void)d_ws; (void)ws_size;
  const float* emis = (const float*)d_in[0];   // (L, M, B, Q)
  const float* A    = (const float*)d_in[1];   // (M, Q, Q)
  const float* init = (const float*)d_in[2];   // (M, Q)
  float* out        = (float*)d_out;           // (M, B)
  // 8 workgroups: one per (model, 16-batch tile); 512 threads = 16 wave32.
  hipLaunchKernelGGL(hmm_forward_kernel, dim3(M_MODELS * (B_BATCH / 16)),
                     dim3(512), 0, stream, emis, A, init, out);
}